// LplsNorm_29910152249796
// MI455X (gfx1250) — compile-verified
//
#include <hip/hip_runtime.h>

typedef __attribute__((ext_vector_type(2))) float v2f;
typedef __attribute__((ext_vector_type(4))) float v4f;
typedef __attribute__((ext_vector_type(8))) float v8f;

#define NDIM 8192
#define ROWS 16          // rows per workgroup band
#define TCOLS 512        // tile width in floats
#define TPAD 516         // padded LDS row stride: (516 % 64) == 4 -> conflict-free b64 A-frag reads
#define NTILES (NDIM / TCOLS)
#define WAVE_COLS (TCOLS / 8)   // 8 waves per block, 64 columns each

// ---------------------------------------------------------------------------
// Pass 1: degree_i = sum_j A[i,j] via V_WMMA_F32_16X16X4_F32 with B == ones,
// then sinv[i] = rsqrt(degree_i).
// One block = 8 waves handles a 16-row band across the full 8192 columns.
// ---------------------------------------------------------------------------
__global__ __launch_bounds__(256)
void lpls_rowsum_rsqrt(const float* __restrict__ A, float* __restrict__ sinv) {
    __shared__ float tile[ROWS][TPAD];
    __shared__ float rowacc[ROWS];

    const int tid  = threadIdx.x;
    const int wave = tid >> 5;
    const int lane = tid & 31;
    const int row0 = blockIdx.x * ROWS;

    if (tid < ROWS) rowacc[tid] = 0.0f;

    v8f c = {};                      // D/C accumulator; every column holds the row-sums
    const v2f bones = {1.0f, 1.0f};  // B = 4x16 all-ones (layout-independent)

    const int lrow  = lane & 15;         // A-matrix row handled by this lane
    const int khalf = (lane >> 4) * 2;   // lanes 0-15: K=0,1 ; lanes 16-31: K=2,3
    const int cbase = wave * WAVE_COLS;  // this wave's 64-column slice of the tile

    for (int t = 0; t < NTILES; ++t) {
        // Coalesced cooperative load: 16 rows x 512 cols = 2048 float4, 8 per thread.
        const v4f* src = (const v4f*)(A + (size_t)row0 * NDIM + (size_t)t * TCOLS);
        #pragma unroll
        for (int it = 0; it < 8; ++it) {
            int f  = it * 256 + tid;   // 0..2047 flat float4 index in tile
            int r  = f >> 7;           // 128 float4 per tile row
            int cq = f & 127;
            v4f v = src[(size_t)r * (NDIM / 4) + cq];
            *(v4f*)&tile[r][cq * 4] = v;   // 16B-aligned ds_store_b128
        }
        __syncthreads();

        // 16 chained WMMA accumulations cover this wave's 16x64 sub-tile.
        #pragma unroll
        for (int k = 0; k < WAVE_COLS; k += 4) {
            v2f a = *(const v2f*)&tile[lrow][cbase + k + khalf];  // ds_load_b64, bank-conflict-free
            c = __builtin_amdgcn_wmma_f32_16x16x4_f32(
                    /*neg_a=*/false, a, /*neg_b=*/false, bones,
                    /*c_mod=*/(short)0, c, /*reuse_a=*/false, /*reuse_b=*/false);
        }
        __syncthreads();
    }

    // Column N=0 of D: lane 0 holds rows 0..7 (VGPR g -> M=g),
    // lane 16 holds rows 8..15 (VGPR g -> M=8+g).
    if (lane == 0) {
        #pragma unroll
        for (int g = 0; g < 8; ++g) atomicAdd(&rowacc[g], c[g]);          // ds_add_f32
    } else if (lane == 16) {
        #pragma unroll
        for (int g = 0; g < 8; ++g) atomicAdd(&rowacc[8 + g], c[g]);
    }
    __syncthreads();

    if (tid < ROWS) {
        sinv[row0 + tid] = rsqrtf(rowacc[tid]);
    }
}

// ---------------------------------------------------------------------------
// Pass 2: out[i,j] = sinv[i] * A[i,j] * sinv[j].
// One block per row; float4 streaming. A read NT (consume L2 residue from
// pass 1 without re-retaining), out stored NT (written once, never re-read).
// sinv (32 KB) read with default policy -> stays hot in cache for all blocks.
// ---------------------------------------------------------------------------
__global__ __launch_bounds__(256)
void lpls_scale(const float* __restrict__ A, const float* __restrict__ sinv,
                float* __restrict__ out) {
    const int i   = blockIdx.x;
    const int tid = threadIdx.x;
    const float si = sinv[i];                         // uniform scalar load

    const v4f* a4 = (const v4f*)(A + (size_t)i * NDIM);
    const v4f* s4 = (const v4f*)sinv;
    v4f*       o4 = (v4f*)(out + (size_t)i * NDIM);

    #pragma unroll
    for (int k = 0; k < 8; ++k) {
        int j = k * 256 + tid;                        // 0..2047 float4 per row
        v4f a = __builtin_nontemporal_load(&a4[j]);   // global_load_b128 ... NT
        v4f s = s4[j];
        v4f r = a * s * si;
        __builtin_nontemporal_store(r, &o4[j]);       // global_store_b128 ... NT
    }
}

extern "C" void kernel_launch(void* const* d_in, const int* in_sizes, int n_in,
                              void* d_out, int out_size, void* d_ws, size_t ws_size,
                              hipStream_t stream) {
    const float* A   = (const float*)d_in[0];
    float*       out = (float*)d_out;
    float*       sinv = (float*)d_ws;   // N floats = 32 KB of scratch

    lpls_rowsum_rsqrt<<<NDIM / ROWS, 256, 0, stream>>>(A, sinv);
    lpls_scale<<<NDIM, 256, 0, stream>>>(A, sinv, out);
}